// ScalarVectorBlock_34892314313407
// MI455X (gfx1250) — compile-verified
//
#include <hip/hip_runtime.h>
#include <math.h>

typedef float v2f __attribute__((ext_vector_type(2)));
typedef float v8f __attribute__((ext_vector_type(8)));
typedef int   v4i __attribute__((ext_vector_type(4)));

#if defined(__has_builtin)
#  if __has_builtin(__builtin_amdgcn_global_load_async_to_lds_b128) && \
      __has_builtin(__builtin_amdgcn_s_wait_asynccnt)
#    define USE_ASYNC_LDS 1
#  endif
#endif
#ifndef USE_ASYNC_LDS
#  define USE_ASYNC_LDS 0
#endif

#if USE_ASYNC_LDS
typedef __attribute__((address_space(3))) v4i as3_v4i;
#endif

__device__ __forceinline__ float gelu_exact(float x) {
    return 0.5f * x * (1.0f + erff(x * 0.70710678118654752440f));
}

// ---------------------------------------------------------------------------
// WMMA f32 GEMM: C[M x Ncols] = act(A[M x K] @ B + bias (+ Cin))
// 4 waves/block, one 16x16 C tile per wave (shared tileM). A is staged into
// LDS in 16x32 chunks via GLOBAL_LOAD_ASYNC_TO_LDS_B128 (ASYNCcnt) and reused
// by all 4 waves; B fragments come straight from global (L2-resident).
// K stepped by 4 with V_WMMA_F32_16X16X4_F32. Batched via blockIdx.z.
// TRANSB: 0 -> B is [K x Ncols] row-major; 1 -> B is [Ncols x K] row-major.
// EPI: 0 = acc+bias; 1 = acc+bias+Cin; 2 = gelu(acc+bias); 3 = sigmoid(acc+bias)
// ---------------------------------------------------------------------------
#define APITCH 36   // floats per A-row in LDS: 16B-aligned, conflict-free reads

template <int TRANSB, int EPI>
__global__ void wmma_gemm(const float* __restrict__ A, const float* __restrict__ Bm,
                          const float* __restrict__ bias, const float* __restrict__ Cin,
                          float* __restrict__ Cout, int Ncols, int K,
                          int lda, int ldb, int ldc,
                          long sA, long sB, long sC) {
    __shared__ float s_a[16 * APITCH];

    const float* Ab = A + (long)blockIdx.z * sA;
    const float* Bb = Bm + (long)blockIdx.z * sB;
    float* Cb = Cout + (long)blockIdx.z * sC;

    int wave = threadIdx.x >> 5;
    int lane = threadIdx.x & 31;
    int tileN = blockIdx.x * 4 + wave;
    int tileM = blockIdx.y;
    bool active = (tileN * 16 < Ncols);   // inactive waves still hit barriers

    int l = lane & 15;
    int hh = lane >> 4;                   // half-wave selects K pair
    int col = tileN * 16 + l;             // B/C column for this lane

    v8f acc = {0.f, 0.f, 0.f, 0.f, 0.f, 0.f, 0.f, 0.f};

    for (int kk = 0; kk < K; kk += 32) {
        __syncthreads();                  // protect prev-chunk LDS reads
        {
            int rr = threadIdx.x >> 3;        // 0..15 (row of A chunk)
            int kq = (threadIdx.x & 7) * 4;   // 0,4,...,28
            const float* gsrc = Ab + (long)(tileM * 16 + rr) * lda + kk + kq;
            float* ldst = &s_a[rr * APITCH + kq];
#if USE_ASYNC_LDS
            __builtin_amdgcn_global_load_async_to_lds_b128((v4i*)gsrc,
                                                           (as3_v4i*)ldst, 0, 0);
#else
            *(float4*)ldst = *(const float4*)gsrc;
#endif
        }
#if USE_ASYNC_LDS
        __builtin_amdgcn_s_wait_asynccnt(0);
#endif
        __syncthreads();
        if (active) {
#pragma unroll
            for (int ks = 0; ks < 32; ks += 4) {
                int k0 = ks + 2 * hh;
                v2f a, b;
                a.x = s_a[l * APITCH + k0];
                a.y = s_a[l * APITCH + k0 + 1];
                int kg = kk + k0;
                if (TRANSB) {
                    b.x = Bb[(long)col * ldb + kg];
                    b.y = Bb[(long)col * ldb + kg + 1];
                } else {
                    b.x = Bb[(long)kg * ldb + col];
                    b.y = Bb[(long)(kg + 1) * ldb + col];
                }
                acc = __builtin_amdgcn_wmma_f32_16x16x4_f32(false, a, false, b,
                                                            (short)0, acc, false, false);
            }
        }
    }

    if (active) {
        float bv = bias ? bias[col] : 0.0f;
#pragma unroll
        for (int r = 0; r < 8; ++r) {
            int orow = tileM * 16 + r + 8 * hh;   // D layout: lanes16-31 -> M=r+8
            float v = acc[r] + bv;
            if (EPI == 1) v += Cin[(long)orow * ldc + col];
            if (EPI == 2) v = gelu_exact(v);
            if (EPI == 3) v = 1.0f / (1.0f + expf(-v));
            Cb[(long)orow * ldc + col] = v;
        }
    }
}

// ---------------------------------------------------------------------------
// Row LayerNorm over D=128: one block (128 threads, 4 waves) per row.
// ---------------------------------------------------------------------------
__global__ void ln_rows(const float* __restrict__ x, const float* __restrict__ g,
                        const float* __restrict__ be, float* __restrict__ y) {
    int row = blockIdx.x, t = threadIdx.x, lane = t & 31, wid = t >> 5;
    __shared__ float sr[4];
    float v = x[(long)row * 128 + t];
    float s = v;
    for (int off = 16; off; off >>= 1) s += __shfl_xor(s, off, 32);
    if (lane == 0) sr[wid] = s;
    __syncthreads();
    float mean = (sr[0] + sr[1] + sr[2] + sr[3]) * (1.0f / 128.0f);
    __syncthreads();
    float d = v - mean;
    float s2 = d * d;
    for (int off = 16; off; off >>= 1) s2 += __shfl_xor(s2, off, 32);
    if (lane == 0) sr[wid] = s2;
    __syncthreads();
    float var = (sr[0] + sr[1] + sr[2] + sr[3]) * (1.0f / 128.0f);
    y[(long)row * 128 + t] = d * rsqrtf(var + 1e-5f) * g[t] + be[t];
}

// ---------------------------------------------------------------------------
// Pair kernel: one block per (b,i). 512 threads (thread = j).
// pass1: h_d = gelu(dist*w_d1[d]+b_d1[d]) kept in REGISTERS (128/thread),
//        logits = (qk + h.qb_i + q_i.b_d2)/sqrt(D), block softmax -> attn
//        (in-place over the Lqk buffer), attn-weighted direction AD[b,i,3].
// pass2: H[b,i,d] = sum_j attn_ij*h_ij[d] via deterministic LDS transpose in
//        8 chunks of 16 channels (pitch 17 -> bank-conflict-free). No second
//        erf pass, no atomics.
// ---------------------------------------------------------------------------
__global__ void __launch_bounds__(512, 1)
pair_kernel(float* __restrict__ attn,        // [B,N,N] in:Lqk out:attn
            const float* __restrict__ q,     // [B,N,D]
            const float* __restrict__ qb,    // [B,N,D]
            const float* __restrict__ coords,// [B,N,3]
            const float* __restrict__ w_d1, const float* __restrict__ b_d1,
            const float* __restrict__ b_d2,
            float* __restrict__ H,           // [B,N,D]
            float* __restrict__ AD) {        // [B,N,3]
    const int Nn = 512, Dd = 128;
    int i = blockIdx.x, b = blockIdx.y;
    int t = threadIdx.x, lane = t & 31, wid = t >> 5;

    __shared__ float s_qb[128], s_wd1[128], s_bd1[128];
    __shared__ float s_red[16];
    __shared__ float s_adw[16][3];
    __shared__ float s_hT[512 * 17];   // chunk transpose buffer (34.8 KB)
    __shared__ float s_p[32 * 17];
    __shared__ float s_qdotb;

    long base = (long)(b * Nn + i);
    if (t < 128) {
        s_qb[t]  = qb[base * Dd + t];
        s_wd1[t] = w_d1[t];
        s_bd1[t] = b_d1[t];
        s_hT[t]  = q[base * Dd + t] * b_d2[t];   // scratch for q . b_d2
    }
    __syncthreads();
    if (t == 0) {
        float s = 0.f;
        for (int d = 0; d < 128; ++d) s += s_hT[d];
        s_qdotb = s;
    }
    float cx = coords[base * 3 + 0];
    float cy = coords[base * 3 + 1];
    float cz = coords[base * 3 + 2];
    __syncthreads();

    // ---- pass 1: thread t = j ----
    long rowbase = (long)b * Nn * Nn + (long)i * Nn;
    float rx = cx - coords[((long)b * Nn + t) * 3 + 0];
    float ry = cy - coords[((long)b * Nn + t) * 3 + 1];
    float rz = cz - coords[((long)b * Nn + t) * 3 + 2];
    float dist = sqrtf(rx * rx + ry * ry + rz * rz);

    float h[128];
    float hdot = 0.f;
#pragma unroll
    for (int d = 0; d < 128; ++d) {
        h[d] = gelu_exact(fmaf(dist, s_wd1[d], s_bd1[d]));
        hdot += h[d] * s_qb[d];
    }

    float logit = (attn[rowbase + t] + hdot + s_qdotb) * 0.08838834764831845f; // 1/sqrt(128)

    // block softmax (16 waves)
    float m = logit;
    for (int off = 16; off; off >>= 1) m = fmaxf(m, __shfl_xor(m, off, 32));
    if (lane == 0) s_red[wid] = m;
    __syncthreads();
    float mm = s_red[0];
    for (int w2 = 1; w2 < 16; ++w2) mm = fmaxf(mm, s_red[w2]);
    __syncthreads();
    float e = expf(logit - mm);
    float ssum = e;
    for (int off = 16; off; off >>= 1) ssum += __shfl_xor(ssum, off, 32);
    if (lane == 0) s_red[wid] = ssum;
    __syncthreads();
    float tot = 0.f;
    for (int w2 = 0; w2 < 16; ++w2) tot += s_red[w2];
    float a = e / tot;
    attn[rowbase + t] = a;

    // attn-weighted direction aggregate
    float invd = 1.0f / fmaxf(dist, 1e-6f);
    float ax = a * rx * invd, ay = a * ry * invd, az = a * rz * invd;
    for (int off = 16; off; off >>= 1) {
        ax += __shfl_xor(ax, off, 32);
        ay += __shfl_xor(ay, off, 32);
        az += __shfl_xor(az, off, 32);
    }
    if (lane == 0) { s_adw[wid][0] = ax; s_adw[wid][1] = ay; s_adw[wid][2] = az; }
    __syncthreads();
    if (t < 3) {
        float s = 0.f;
        for (int w2 = 0; w2 < 16; ++w2) s += s_adw[w2][t];
        AD[base * 3 + t] = s;
    }

    // ---- pass 2: chunked LDS transpose, 16 channels at a time ----
#pragma unroll
    for (int c = 0; c < 8; ++c) {
        __syncthreads();                       // protect prev chunk's reads
#pragma unroll
        for (int k2 = 0; k2 < 16; ++k2)
            s_hT[t * 17 + k2] = h[c * 16 + k2] * a;   // pre-weighted by attn
        __syncthreads();
        int d2 = t & 15, g = t >> 4;           // 32 j-groups x 16 channels
        float p = 0.f;
#pragma unroll
        for (int jj = 0; jj < 16; ++jj)
            p += s_hT[(g * 16 + jj) * 17 + d2];
        s_p[g * 17 + d2] = p;
        __syncthreads();
        if (t < 16) {
            float s = 0.f;
            for (int g2 = 0; g2 < 32; ++g2) s += s_p[g2 * 17 + t];
            H[base * Dd + c * 16 + t] = s;
        }
    }
}

// ---------------------------------------------------------------------------
// Final vector path: one block per (b,i), 48 threads (c = t/16, v = t%16).
// agg = aggv + AD[c]*gate[v]; LN over v (width-16 shuffles); @ w_vo; + vector
// ---------------------------------------------------------------------------
__global__ void vecout_kernel(const float* __restrict__ aggv, const float* __restrict__ AD,
                              const float* __restrict__ gate, const float* __restrict__ vec,
                              const float* __restrict__ g_v, const float* __restrict__ be_v,
                              const float* __restrict__ w_vo, const float* __restrict__ b_vo,
                              float* __restrict__ out) {
    int idx = blockIdx.x, t = threadIdx.x;
    int c = t >> 4, vl = t & 15;
    __shared__ float s_ln[48];
    float a = aggv[(long)idx * 48 + t] + AD[(long)idx * 3 + c] * gate[(long)idx * 16 + vl];
    float s = a;
    for (int off = 8; off; off >>= 1) s += __shfl_xor(s, off, 16);
    float mean = s * (1.0f / 16.0f);
    float d = a - mean;
    float s2 = d * d;
    for (int off = 8; off; off >>= 1) s2 += __shfl_xor(s2, off, 16);
    float var = s2 * (1.0f / 16.0f);
    s_ln[t] = d * rsqrtf(var + 1e-5f) * g_v[vl] + be_v[vl];
    __syncthreads();
    float o = b_vo[vl];
#pragma unroll
    for (int u = 0; u < 16; ++u) o += s_ln[c * 16 + u] * w_vo[u * 16 + vl];
    out[(long)idx * 48 + t] = vec[(long)idx * 48 + t] + o;
}

extern "C" void kernel_launch(void* const* d_in, const int* in_sizes, int n_in,
                              void* d_out, int out_size, void* d_ws, size_t ws_size,
                              hipStream_t stream) {
    (void)in_sizes; (void)n_in; (void)out_size; (void)ws_size;
    const int B = 2, N = 512;

    const float* scalar = (const float*)d_in[0];
    const float* vector = (const float*)d_in[1];
    const float* coords = (const float*)d_in[2];
    const float* w_q  = (const float*)d_in[3];  const float* b_q  = (const float*)d_in[4];
    const float* w_k  = (const float*)d_in[5];  const float* b_k  = (const float*)d_in[6];
    const float* w_v  = (const float*)d_in[7];  const float* b_v  = (const float*)d_in[8];
    const float* w_d1 = (const float*)d_in[9];  const float* b_d1 = (const float*)d_in[10];
    const float* w_d2 = (const float*)d_in[11]; const float* b_d2 = (const float*)d_in[12];
    const float* w_g  = (const float*)d_in[13]; const float* b_g  = (const float*)d_in[14];
    const float* w_o  = (const float*)d_in[15]; const float* b_o  = (const float*)d_in[16];
    const float* w_f1 = (const float*)d_in[17]; const float* b_f1 = (const float*)d_in[18];
    const float* w_f2 = (const float*)d_in[19]; const float* b_f2 = (const float*)d_in[20];
    const float* w_vo = (const float*)d_in[21]; const float* b_vo = (const float*)d_in[22];
    const float* g_s  = (const float*)d_in[23]; const float* be_s = (const float*)d_in[24];
    const float* g_v  = (const float*)d_in[25]; const float* be_v = (const float*)d_in[26];

    // workspace layout (floats)
    float* ws   = (float*)d_ws;
    float* sn   = ws;               // 131072
    float* q    = sn   + 131072;    // 131072
    float* k    = q    + 131072;    // 131072
    float* vbuf = k    + 131072;    // 131072
    float* qb   = vbuf + 131072;    // 131072
    float* attn = qb   + 131072;    // 524288 (Lqk then attn, in place)
    float* H    = attn + 524288;    // 131072
    float* U    = H    + 131072;    // 131072
    float* S1   = U    + 131072;    // 131072
    float* h1   = S1   + 131072;    // 131072
    float* F    = h1   + 131072;    // 262144
    float* gate = F    + 262144;    // 16384
    float* aggv = gate + 16384;     // 49152
    float* AD   = aggv + 49152;     // 3072

    float* outS = (float*)d_out;        // [B,N,128] final scalar
    float* outV = outS + B * N * 128;   // [B,N,3,16] final vector

    dim3 blk(128);

    // 1) sn = LN(scalar)
    ln_rows<<<B * N, 128, 0, stream>>>(scalar, g_s, be_s, sn);

    // 2) q,k,v = sn @ W + b   (M = B*N = 1024)
    wmma_gemm<0,0><<<dim3(2,64,1), blk, 0, stream>>>(sn, w_q, b_q, nullptr, q,    128,128, 128,128,128, 0,0,0);
    wmma_gemm<0,0><<<dim3(2,64,1), blk, 0, stream>>>(sn, w_k, b_k, nullptr, k,    128,128, 128,128,128, 0,0,0);
    wmma_gemm<0,0><<<dim3(2,64,1), blk, 0, stream>>>(sn, w_v, b_v, nullptr, vbuf, 128,128, 128,128,128, 0,0,0);

    // 3) qb = q @ w_d2^T
    wmma_gemm<1,0><<<dim3(2,64,1), blk, 0, stream>>>(q, w_d2, nullptr, nullptr, qb, 128,128, 128,128,128, 0,0,0);

    // 4) Lqk = q @ k^T  (per batch)
    wmma_gemm<1,0><<<dim3(8,32,2), blk, 0, stream>>>(q, k, nullptr, nullptr, attn, 512,128, 128,128,512,
                                                     65536, 65536, 262144);

    // 5) pair loop: softmax + H + AD (attn in place)
    pair_kernel<<<dim3(N, B), 512, 0, stream>>>(attn, q, qb, coords, w_d1, b_d1, b_d2, H, AD);

    // 6) U = attn @ v   (per batch)
    wmma_gemm<0,0><<<dim3(2,32,2), blk, 0, stream>>>(attn, vbuf, nullptr, nullptr, U, 128,512, 512,128,128,
                                                     262144, 65536, 65536);

    // 7) U = U + H @ w_d2 + b_d2
    wmma_gemm<0,1><<<dim3(2,64,1), blk, 0, stream>>>(H, w_d2, b_d2, U, U, 128,128, 128,128,128, 0,0,0);

    // 8) S1 = scalar + U @ w_o + b_o
    wmma_gemm<0,1><<<dim3(2,64,1), blk, 0, stream>>>(U, w_o, b_o, scalar, S1, 128,128, 128,128,128, 0,0,0);

    // 9) h1 = LN(S1)
    ln_rows<<<B * N, 128, 0, stream>>>(S1, g_s, be_s, h1);

    // 10) F = gelu(h1 @ w_f1 + b_f1)
    wmma_gemm<0,2><<<dim3(4,64,1), blk, 0, stream>>>(h1, w_f1, b_f1, nullptr, F, 256,128, 128,256,256, 0,0,0);

    // 11) S2(out) = S1 + F @ w_f2 + b_f2
    wmma_gemm<0,1><<<dim3(2,64,1), blk, 0, stream>>>(F, w_f2, b_f2, S1, outS, 128,256, 256,128,128, 0,0,0);

    // 12) gate = sigmoid(S2 @ w_g + b_g)
    wmma_gemm<0,3><<<dim3(1,64,1), blk, 0, stream>>>(outS, w_g, b_g, nullptr, gate, 16,128, 128,16,16, 0,0,0);

    // 13) aggv = attn @ vector_flat  (per batch, 48 cols)
    wmma_gemm<0,0><<<dim3(1,32,2), blk, 0, stream>>>(attn, vector, nullptr, nullptr, aggv, 48,512, 512,48,48,
                                                     262144, 24576, 24576);

    // 14) final vector path
    vecout_kernel<<<B * N, 48, 0, stream>>>(aggv, AD, gate, vector, g_v, be_v, w_vo, b_vo, outV);
}